// GCN_3642132267778
// MI455X (gfx1250) — compile-verified
//
#include <hip/hip_runtime.h>

typedef __attribute__((ext_vector_type(2))) float v2f;
typedef __attribute__((ext_vector_type(8))) float v8f;

#define F_OUT 64   // hidden == out_feats == 64 for both layers

// ---------------- utility kernels ----------------

__global__ void zero_kernel(float* __restrict__ p, long n) {
    long i = (long)blockIdx.x * blockDim.x + threadIdx.x;
    long stride = (long)gridDim.x * blockDim.x;
    for (; i < n; i += stride) p[i] = 0.0f;
}

__global__ void degree_kernel(const int* __restrict__ src, const int* __restrict__ dst,
                              float* __restrict__ deg_out, float* __restrict__ deg_in, int E) {
    int e = blockIdx.x * blockDim.x + threadIdx.x;
    if (e < E) {
        atomicAdd(&deg_out[src[e]], 1.0f);
        atomicAdd(&deg_in[dst[e]], 1.0f);
    }
}

__global__ void norm_kernel(const float* __restrict__ deg_out, const float* __restrict__ deg_in,
                            float* __restrict__ norm_out, float* __restrict__ norm_in, int N) {
    int i = blockIdx.x * blockDim.x + threadIdx.x;
    if (i < N) {
        norm_out[i] = rsqrtf(fmaxf(deg_out[i], 1.0f));
        norm_in[i]  = rsqrtf(fmaxf(deg_in[i], 1.0f));
    }
}

// ---------------- WMMA GEMM:  H[nrows, 64] = (X * scale) @ W[KD, 64] ----------------
// Block = 128 threads = 4 waves; each wave computes a 16x64 tile (4x v_wmma_f32_16x16x4_f32
// accumulators), K-loop in steps of 4. W is staged in LDS once per block.
// Full-tile waves store unguarded; only the boundary tile takes per-element guards.
template <int KD>
__global__ __launch_bounds__(128) void gemm_wmma(const float* __restrict__ X,
                                                 const float* __restrict__ W,
                                                 const float* __restrict__ scale,  // may be null
                                                 float* __restrict__ Hout, int nrows) {
    __shared__ float wlds[KD * F_OUT];
    int tid = threadIdx.x;
    for (int i = tid; i < KD * F_OUT; i += 128) wlds[i] = W[i];
    __syncthreads();

    int wave = tid >> 5;
    int lane = tid & 31;
    int l15  = lane & 15;
    int hi   = lane >> 4;          // 0 for lanes 0-15, 1 for lanes 16-31
    int koff = hi * 2;             // A/B fragment K offset per ISA layout

    int row0 = blockIdx.x * 64 + wave * 16;
    int rA = row0 + l15;
    if (rA >= nrows) rA = nrows - 1;   // clamp reads; stores handled below
    float s = scale ? scale[rA] : 1.0f;

    const float* xrow = X + (long)rA * KD;

    v8f acc[4] = {};                   // four 16x16 f32 accumulators (N tiles 0..3)

    for (int k = 0; k < KD; k += 4) {
        v2f a;
        a.x = xrow[k + koff]     * s;
        a.y = xrow[k + koff + 1] * s;
        __builtin_prefetch(xrow + k + 16, 0, 1);   // global_prefetch_b8 ahead in the row
#pragma unroll
        for (int n = 0; n < 4; ++n) {
            v2f b;
            b.x = wlds[(k + koff)     * F_OUT + n * 16 + l15];
            b.y = wlds[(k + koff + 1) * F_OUT + n * 16 + l15];
            // 8 args: (neg_a, A, neg_b, B, c_mod, C, reuse_a, reuse_b)
            acc[n] = __builtin_amdgcn_wmma_f32_16x16x4_f32(
                false, a, false, b, (short)0, acc[n], false, false);
        }
    }

    // C/D layout: VGPR r -> row M=r (lanes 0-15) / M=r+8 (lanes 16-31), col N = lane&15 (+16n)
    float* outp = Hout + (long)(row0 + hi * 8) * F_OUT + l15;
    if (row0 + 16 <= nrows) {
        // fast path: whole 16-row tile in range, no per-element guards
#pragma unroll
        for (int r = 0; r < 8; ++r) {
#pragma unroll
            for (int n = 0; n < 4; ++n) outp[(long)r * F_OUT + n * 16] = acc[n][r];
        }
    } else {
#pragma unroll
        for (int r = 0; r < 8; ++r) {
            int row = row0 + r + hi * 8;
            if (row < nrows) {
#pragma unroll
                for (int n = 0; n < 4; ++n) outp[(long)r * F_OUT + n * 16] = acc[n][r];
            }
        }
    }
}

// ---------------- edge scatter-add:  agg[dst] += H[src]  (per feature) ----------------
__global__ void scatter_kernel(const float* __restrict__ H, const int* __restrict__ src,
                               const int* __restrict__ dst, float* __restrict__ agg, long total) {
    long idx = (long)blockIdx.x * blockDim.x + threadIdx.x;
    if (idx >= total) return;
    int e = (int)(idx >> 6);
    int f = (int)(idx & 63);
    int si = __ldg(src + e);
    int di = __ldg(dst + e);
    atomicAdd(&agg[(long)di * F_OUT + f], H[(long)si * F_OUT + f]);
}

// ---------------- epilogue 1: h1s = relu(agg*norm_in + b1) * norm_out ----------------
__global__ void act_kernel(const float* __restrict__ agg, const float* __restrict__ norm_in,
                           const float* __restrict__ norm_out, const float* __restrict__ b,
                           float* __restrict__ out, long total) {
    long idx = (long)blockIdx.x * blockDim.x + threadIdx.x;
    if (idx >= total) return;
    int i = (int)(idx >> 6);
    int f = (int)(idx & 63);
    float v = agg[idx] * norm_in[i] + b[f];
    out[idx] = fmaxf(v, 0.0f) * norm_out[i];
}

// ---------------- epilogue 2 (in place on d_out): out = out*norm_in + b2 ----------------
__global__ void final_kernel(float* __restrict__ out, const float* __restrict__ norm_in,
                             const float* __restrict__ b, long total) {
    long idx = (long)blockIdx.x * blockDim.x + threadIdx.x;
    if (idx >= total) return;
    int i = (int)(idx >> 6);
    int f = (int)(idx & 63);
    out[idx] = out[idx] * norm_in[i] + b[f];
}

extern "C" void kernel_launch(void* const* d_in, const int* in_sizes, int n_in,
                              void* d_out, int out_size, void* d_ws, size_t ws_size,
                              hipStream_t stream) {
    const float* x   = (const float*)d_in[0];   // [N,128]
    const int*   src = (const int*)d_in[1];     // [E]
    const int*   dst = (const int*)d_in[2];     // [E]
    const float* W1  = (const float*)d_in[3];   // [128,64]
    const float* b1  = (const float*)d_in[4];   // [64]
    const float* W2  = (const float*)d_in[5];   // [64,64]
    const float* b2  = (const float*)d_in[6];   // [64]
    float* out = (float*)d_out;                 // [N,64]

    const int N = in_sizes[0] / 128;
    const int E = in_sizes[1];
    const long NF = (long)N * F_OUT;
    const long EF = (long)E * F_OUT;

    // workspace layout (floats)
    float* w        = (float*)d_ws;
    float* deg_out  = w;                // N
    float* deg_in   = w + (long)N;      // N
    float* norm_out = w + 2L * N;       // N
    float* norm_in  = w + 3L * N;       // N
    float* bufA     = w + 4L * N;       // N*64  (h, then h1s)
    float* bufB     = bufA + NF;        // N*64  (agg1, then h2)

    const int T = 256;

    // 1) zero: degrees (2N contiguous), agg1, and d_out (scatter target for layer 2)
    zero_kernel<<<1024, T, 0, stream>>>(deg_out, 2L * N);
    zero_kernel<<<2048, T, 0, stream>>>(bufB, NF);
    zero_kernel<<<2048, T, 0, stream>>>(out, NF);

    // 2) degrees + norms
    degree_kernel<<<(E + T - 1) / T, T, 0, stream>>>(src, dst, deg_out, deg_in, E);
    norm_kernel<<<(N + T - 1) / T, T, 0, stream>>>(deg_out, deg_in, norm_out, norm_in, N);

    // 3) layer 1 GEMM (WMMA fp32): bufA = (x * norm_out) @ W1
    int gemm_blocks = (N + 63) / 64;
    gemm_wmma<128><<<gemm_blocks, 128, 0, stream>>>(x, W1, norm_out, bufA, N);

    // 4) scatter: bufB[dst] += bufA[src]
    scatter_kernel<<<(int)((EF + T - 1) / T), T, 0, stream>>>(bufA, src, dst, bufB, EF);

    // 5) epilogue 1 (fuses relu + b1 + pre-scale by norm_out for layer 2): bufA = ...
    act_kernel<<<(int)((NF + T - 1) / T), T, 0, stream>>>(bufB, norm_in, norm_out, b1, bufA, NF);

    // 6) layer 2 GEMM (WMMA fp32): bufB = bufA @ W2
    gemm_wmma<64><<<gemm_blocks, 128, 0, stream>>>(bufA, W2, nullptr, bufB, N);

    // 7) scatter into output, then final scale + bias in place
    scatter_kernel<<<(int)((EF + T - 1) / T), T, 0, stream>>>(bufB, src, dst, out, EF);
    final_kernel<<<(int)((NF + T - 1) / T), T, 0, stream>>>(out, norm_in, b2, NF);
}